// KmerGCNEncoder_71150428225575
// MI455X (gfx1250) — compile-verified
//
#include <hip/hip_runtime.h>
#include <hip/hip_bf16.h>

// ---------------------------------------------------------------------------
// KmerGCNEncoder for MI455X (gfx1250, wave32).
// Pipeline: emb gather -> GEMM(W1) [WMMA f32 16x16x4] -> edge scatter-sum
//           -> +selfloop+bias+ReLU -> GEMM(W2) -> scatter -> +selfloop+bias
//           -> mean pool per graph.
// ---------------------------------------------------------------------------

typedef __attribute__((ext_vector_type(2))) float v2f;
typedef __attribute__((ext_vector_type(8))) float v8f;

#define EMB 128
#define HID 128
#define WAVES_PER_BLOCK 2   // 2 waves/block -> 32 rows/block, N=100000 -> 3125 blocks

// ---------------- degree / normalization ----------------

__global__ void deg_init_kernel(float* __restrict__ deg, int n) {
    int i = blockIdx.x * blockDim.x + threadIdx.x;
    if (i < n) deg[i] = 1.0f;                     // self-loop contribution
}

__global__ void deg_accum_kernel(const int* __restrict__ dst,
                                 float* __restrict__ deg, int n_edges) {
    int e = blockIdx.x * blockDim.x + threadIdx.x;
    if (e < n_edges) atomicAdd(&deg[dst[e]], 1.0f);
}

__global__ void deg_finish_kernel(float* __restrict__ deg, int n) {
    int i = blockIdx.x * blockDim.x + threadIdx.x;
    if (i < n) deg[i] = rsqrtf(deg[i]);           // deg >= 1 always
}

// ---------------- generic zero fill ----------------

__global__ void zero_kernel(float* __restrict__ p, int n) {
    int i = blockIdx.x * blockDim.x + threadIdx.x;
    if (i < n) p[i] = 0.0f;
}

// ---------------- fused (gather +) GEMM: out[r,:] = in[idx(r),:] @ W --------
// One wave handles a 16-row tile. A tile staged in LDS; B streamed from the
// L2-resident 128x128 weight matrix. WMMA F32 16x16x4, 8 N-tiles of 16 cols.

__global__ __launch_bounds__(WAVES_PER_BLOCK * 32)
void gemm_rows_kernel(const float* __restrict__ in,   // [*,128]
                      const int*   __restrict__ idx,  // row map (or nullptr)
                      const float* __restrict__ W,    // [128,128] row-major
                      float*       __restrict__ out,  // [n,128]
                      int n) {
    __shared__ float sA[WAVES_PER_BLOCK][16 * EMB];   // 16 KB

    const int wave = threadIdx.x >> 5;
    const int lane = threadIdx.x & 31;
    const int row0 = (blockIdx.x * WAVES_PER_BLOCK + wave) * 16;
    const bool active = (row0 < n);                   // wave-uniform

    if (active) {
        float* a = sA[wave];
        // stage 16 rows (512 B each) into LDS; one float4 per lane per row
        for (int r = 0; r < 16; ++r) {
            int node = row0 + r;
            const float* srcRow =
                idx ? (in + (size_t)idx[node] * EMB) : (in + (size_t)node * EMB);
            float4 v = ((const float4*)srcRow)[lane];
            ((float4*)(a + r * EMB))[lane] = v;
        }
    }
    __syncthreads();
    if (!active) return;

    const float* a = sA[wave];
    const int arow = lane & 15;           // M index of this lane's A rows
    const int koff = (lane >> 4) * 2;     // lanes 0-15: K{0,1}; 16-31: K{2,3}

    v8f acc[8];
    #pragma unroll
    for (int t = 0; t < 8; ++t) acc[t] = (v8f){};

    for (int ks = 0; ks < EMB; ks += 4) {
        v2f af;
        af.x = a[arow * EMB + ks + koff];
        af.y = a[arow * EMB + ks + koff + 1];
        const float* Wk = W + (size_t)(ks + koff) * HID;   // rows ks+koff, +1
        #pragma unroll
        for (int t = 0; t < 8; ++t) {
            const int col = t * 16 + (lane & 15);
            v2f bf;
            bf.x = Wk[col];
            bf.y = Wk[HID + col];
            // D = A(16x4) x B(4x16) + C, fp32 all the way
            acc[t] = __builtin_amdgcn_wmma_f32_16x16x4_f32(
                false, af, false, bf, (short)0, acc[t], false, false);
        }
    }

    // C/D layout: VGPR r -> M = r + (lane>=16 ? 8 : 0), N = lane & 15
    float* orow = out + (size_t)row0 * HID;
    const int ccol  = lane & 15;
    const int rbase = (lane >> 4) * 8;
    #pragma unroll
    for (int t = 0; t < 8; ++t)
        #pragma unroll
        for (int r = 0; r < 8; ++r)
            orow[(size_t)(rbase + r) * HID + t * 16 + ccol] = acc[t][r];
}

// ---------------- edge scatter: out[dst] += in[src] * dis[src]*dis[dst] ----
// One wave per edge; each lane moves 4 contiguous floats of the 128-wide row.

__global__ void scatter_kernel(const float* __restrict__ in,
                               float*       __restrict__ out,
                               const int*   __restrict__ src,
                               const int*   __restrict__ dst,
                               const float* __restrict__ dis,
                               int n_edges) {
    int e    = blockIdx.x * (blockDim.x >> 5) + (threadIdx.x >> 5);
    int lane = threadIdx.x & 31;
    if (e >= n_edges) return;
    int s = src[e], d = dst[e];
    float w = dis[s] * dis[d];
    float4 v = ((const float4*)(in + (size_t)s * HID))[lane];
    float* o = out + (size_t)d * HID + lane * 4;
    atomicAdd(o + 0, v.x * w);
    atomicAdd(o + 1, v.y * w);
    atomicAdd(o + 2, v.z * w);
    atomicAdd(o + 3, v.w * w);
}

// ---------------- self-loop + bias (+ ReLU) --------------------------------

__global__ void finalize_kernel(float*       __restrict__ scat,  // in/out [n,128]
                                const float* __restrict__ hW,    // [n,128]
                                const float* __restrict__ dis,   // [n]
                                const float* __restrict__ bias,  // [128]
                                int n, int do_relu) {
    int i = blockIdx.x * blockDim.x + threadIdx.x;
    if (i >= n * HID) return;
    int node = i >> 7;
    int c    = i & 127;
    float s  = dis[node];
    float v  = scat[i] + hW[i] * (s * s) + bias[c];
    if (do_relu) v = fmaxf(v, 0.0f);
    scat[i] = v;
}

// ---------------- global mean pool -----------------------------------------

__global__ void pool_accum_kernel(const float* __restrict__ h,
                                  const int*   __restrict__ batch,
                                  float* __restrict__ sums,   // [G,128]
                                  float* __restrict__ cnts,   // [G]
                                  int n) {
    int node = blockIdx.x * (blockDim.x >> 5) + (threadIdx.x >> 5);
    int lane = threadIdx.x & 31;
    if (node >= n) return;
    int g = batch[node];
    float4 v = ((const float4*)(h + (size_t)node * HID))[lane];
    float* o = sums + (size_t)g * HID + lane * 4;
    atomicAdd(o + 0, v.x);
    atomicAdd(o + 1, v.y);
    atomicAdd(o + 2, v.z);
    atomicAdd(o + 3, v.w);
    if (lane == 0) atomicAdd(&cnts[g], 1.0f);
}

__global__ void pool_finish_kernel(const float* __restrict__ sums,
                                   const float* __restrict__ cnts,
                                   float* __restrict__ out, int total) {
    int i = blockIdx.x * blockDim.x + threadIdx.x;
    if (i >= total) return;
    out[i] = sums[i] / fmaxf(cnts[i >> 7], 1.0f);
}

// ---------------------------------------------------------------------------

extern "C" void kernel_launch(void* const* d_in, const int* in_sizes, int n_in,
                              void* d_out, int out_size, void* d_ws, size_t ws_size,
                              hipStream_t stream) {
    const int*   x     = (const int*)  d_in[0];
    const int*   ei    = (const int*)  d_in[1];
    const int*   batch = (const int*)  d_in[2];
    const float* emb   = (const float*)d_in[3];
    const float* W1    = (const float*)d_in[4];
    const float* b1    = (const float*)d_in[5];
    const float* W2    = (const float*)d_in[6];
    const float* b2    = (const float*)d_in[7];
    float* out = (float*)d_out;

    const int N = in_sizes[0];
    const int E = in_sizes[1] / 2;
    const int G = out_size / HID;

    const int* src = ei;
    const int* dst = ei + E;

    // workspace layout
    char* ws = (char*)d_ws;
    float* dis  = (float*)ws;  ws += (((size_t)N * 4) + 255) & ~(size_t)255;
    float* bufA = (float*)ws;  ws += (size_t)N * HID * 4;      // hW (GEMM out)
    float* bufB = (float*)ws;  ws += (size_t)N * HID * 4;      // scatter target
    float* psum = (float*)ws;  ws += (size_t)G * HID * 4;
    float* pcnt = (float*)ws;

    const int T = 256;
    const int nTileBlocks = (N + 16 * WAVES_PER_BLOCK - 1) / (16 * WAVES_PER_BLOCK);
    const int edgeBlocks  = (E + (T / 32) - 1) / (T / 32);
    const int nodeBlocks  = (N + (T / 32) - 1) / (T / 32);
    const int nhBlocks    = (N * HID + T - 1) / T;

    // 1) symmetric-normalization coefficients dis = rsqrt(deg)  (deg incl. self-loop)
    deg_init_kernel  <<<(N + T - 1) / T, T, 0, stream>>>(dis, N);
    deg_accum_kernel <<<(E + T - 1) / T, T, 0, stream>>>(dst, dis, E);
    deg_finish_kernel<<<(N + T - 1) / T, T, 0, stream>>>(dis, N);

    // 2) layer 1: bufA = emb[x] @ W1
    gemm_rows_kernel<<<nTileBlocks, WAVES_PER_BLOCK * 32, 0, stream>>>(
        emb, x, W1, bufA, N);
    //    bufB = scatter(bufA);  then bufB = relu(bufB + bufA*dis^2 + b1)
    zero_kernel    <<<nhBlocks, T, 0, stream>>>(bufB, N * HID);
    scatter_kernel <<<edgeBlocks, T, 0, stream>>>(bufA, bufB, src, dst, dis, E);
    finalize_kernel<<<nhBlocks, T, 0, stream>>>(bufB, bufA, dis, b1, N, 1);

    // 3) layer 2: bufA = bufB @ W2
    gemm_rows_kernel<<<nTileBlocks, WAVES_PER_BLOCK * 32, 0, stream>>>(
        bufB, nullptr, W2, bufA, N);
    zero_kernel    <<<nhBlocks, T, 0, stream>>>(bufB, N * HID);
    scatter_kernel <<<edgeBlocks, T, 0, stream>>>(bufA, bufB, src, dst, dis, E);
    finalize_kernel<<<nhBlocks, T, 0, stream>>>(bufB, bufA, dis, b2, N, 0);

    // 4) mean pool
    zero_kernel<<<(G * HID + G + T - 1) / T, T, 0, stream>>>(psum, G * HID + G);
    pool_accum_kernel <<<nodeBlocks, T, 0, stream>>>(bufB, batch, psum, pcnt, N);
    pool_finish_kernel<<<(G * HID + T - 1) / T, T, 0, stream>>>(psum, pcnt, out, G * HID);
}